// GraphNet_73976516706508
// MI455X (gfx1250) — compile-verified
//
#include <hip/hip_runtime.h>
#include <math.h>

#define N_CONT   99996
#define N_NODES  100000
#define DFEAT    128
#define E_EDGES  1600000
#define BATCH    4096
#define NCLS     10
#define CAT_LEN  10
#define N_CAT    4
#define SP_BETA  1.1f

typedef __attribute__((ext_vector_type(2))) float v2f;
typedef __attribute__((ext_vector_type(8))) float v8f;

// ---------------------------------------------------------------------------
// 0) init: deg=1 (self loop weight), cat_feats einsum, pooled=0
// ---------------------------------------------------------------------------
__global__ void k_init(const float* __restrict__ cat_x,
                       const float* __restrict__ emb_w,
                       float* __restrict__ deg,
                       float* __restrict__ catf,
                       float* __restrict__ pooled) {
  int idx = blockIdx.x * blockDim.x + threadIdx.x;
  if (idx < N_NODES) deg[idx] = 1.0f;             // self-loop contribution
  if (idx < N_CAT * DFEAT) {                      // cat_feats[k][d] = sum_c cat_x[k,d,c]*emb_w[k,c]
    int k = idx / DFEAT, d = idx % DFEAT;
    float s = 0.f;
#pragma unroll
    for (int c = 0; c < CAT_LEN; ++c)
      s += cat_x[(k * DFEAT + d) * CAT_LEN + c] * emb_w[k * CAT_LEN + c];
    catf[idx] = s;
  }
  if (idx < DFEAT) pooled[idx] = 0.f;
}

// ---------------------------------------------------------------------------
// 1) scatter edge weights into degree (col side)
// ---------------------------------------------------------------------------
__global__ void k_deg(const int* __restrict__ col,
                      const float* __restrict__ ew,
                      float* __restrict__ deg) {
  int e = blockIdx.x * blockDim.x + threadIdx.x;
  if (e < E_EDGES) atomicAdd(&deg[col[e]], ew[e]);
}

// ---------------------------------------------------------------------------
// 2) deg -> dis = rsqrt(deg) (in place)
// ---------------------------------------------------------------------------
__global__ void k_dis(float* __restrict__ deg) {
  int i = blockIdx.x * blockDim.x + threadIdx.x;
  if (i < N_NODES) {
    float d = deg[i];
    deg[i] = d > 0.f ? rsqrtf(fmaxf(d, 1e-30f)) : 0.f;
  }
}

// ---------------------------------------------------------------------------
// 3) xw = x @ conv_w via V_WMMA_F32_16X16X4_F32.
//    One block per 16-row strip (6250 strips), 8 waves/block, wave w owns
//    column tile w. A strip staged in LDS (coalesced), B read from global
//    (64KB weights stay L2/L1 resident). EXEC all-ones (no divergence).
// ---------------------------------------------------------------------------
__global__ __launch_bounds__(256)
void k_gemm(const float* __restrict__ num_x,
            const float* __restrict__ catf,
            const float* __restrict__ w,
            float* __restrict__ xw) {
  __shared__ float As[16 * DFEAT];                // 8 KB
  const int tile = blockIdx.x;                    // row tile
  const int tid  = threadIdx.x;

  for (int i = tid; i < 16 * DFEAT; i += 256) {   // coalesced A load
    int r = i >> 7, c = i & 127;
    int row = tile * 16 + r;
    As[i] = (row < N_CONT) ? num_x[row * DFEAT + c]
                           : catf[(row - N_CONT) * DFEAT + c];
  }
  __syncthreads();

  const int wave = tid >> 5;                      // 0..7 -> column tile
  const int lane = tid & 31;
  const int half = lane >> 4;                     // 0: K/M group 0, 1: group 1
  const int l    = lane & 15;
  const int n    = wave * 16 + l;                 // output column

  v8f acc = {};
#pragma unroll
  for (int k0 = 0; k0 < DFEAT; k0 += 4) {
    const int ka = k0 + 2 * half;
    v2f a; a.x = As[l * DFEAT + ka];  a.y = As[l * DFEAT + ka + 1];
    v2f b; b.x = w[ka * DFEAT + n];   b.y = w[(ka + 1) * DFEAT + n];
    acc = __builtin_amdgcn_wmma_f32_16x16x4_f32(
        /*neg_a=*/false, a, /*neg_b=*/false, b,
        /*c_mod=*/(short)0, acc, /*reuse_a=*/false, /*reuse_b=*/false);
  }

  const int row0 = tile * 16 + half * 8;          // C layout: vgpr r -> M = r + 8*half
#pragma unroll
  for (int r = 0; r < 8; ++r)
    xw[(row0 + r) * DFEAT + n] = acc[r];
}

// ---------------------------------------------------------------------------
// 4) self-loop message initializes h: h[i] = xw[i] * dis[i]^2
// ---------------------------------------------------------------------------
__global__ void k_self(const float* __restrict__ xw,
                       const float* __restrict__ dis,
                       float* __restrict__ h) {
  int stride = gridDim.x * blockDim.x;
  for (int idx = blockIdx.x * blockDim.x + threadIdx.x;
       idx < N_NODES * DFEAT; idx += stride) {
    float s = dis[idx >> 7];
    h[idx] = xw[idx] * s * s;
  }
}

// ---------------------------------------------------------------------------
// 5) edge messages: one wave32 per edge; float4 gather from xw[row],
//    4 fp32 global atomics per lane into h[col]. xw & h are L2-resident.
// ---------------------------------------------------------------------------
__global__ __launch_bounds__(256)
void k_edges(const int* __restrict__ row_idx,
             const int* __restrict__ col_idx,
             const float* __restrict__ ew,
             const float* __restrict__ dis,
             const float* __restrict__ xw,
             float* __restrict__ h) {
  const int wave = (blockIdx.x * blockDim.x + threadIdx.x) >> 5;
  const int lane = threadIdx.x & 31;
  const int nwaves = (gridDim.x * blockDim.x) >> 5;
  for (int e = wave; e < E_EDGES; e += nwaves) {
    const int r = row_idx[e];
    const int c = col_idx[e];
    const float nrm = dis[r] * ew[e] * dis[c];
    const float4 v = *(const float4*)(xw + r * DFEAT + lane * 4);
    float* hp = h + c * DFEAT + lane * 4;
    atomicAdd(hp + 0, v.x * nrm);
    atomicAdd(hp + 1, v.y * nrm);
    atomicAdd(hp + 2, v.z * nrm);
    atomicAdd(hp + 3, v.w * nrm);
  }
}

// ---------------------------------------------------------------------------
// 6) pooled[col] = sum_rows relu(h) ; column fixed per thread, 1 atomic/thread
// ---------------------------------------------------------------------------
__global__ void k_pool(const float* __restrict__ h,
                       float* __restrict__ pooled) {
  const int gid = blockIdx.x * blockDim.x + threadIdx.x;
  const int T = gridDim.x * blockDim.x;
  const int col = gid & 127;
  const int rstep = T >> 7;
  float s = 0.f;
  for (int row = gid >> 7; row < N_NODES; row += rstep)
    s += fmaxf(h[row * DFEAT + col], 0.f);
  atomicAdd(&pooled[col], s);
}

// ---------------------------------------------------------------------------
// 7) head: logit = [vanilla, pooled_mean] @ fc_w + fc_b ; out = vanilla / t
// ---------------------------------------------------------------------------
__global__ void k_final(const float* __restrict__ vanilla,
                        const float* __restrict__ fc_w,
                        const float* __restrict__ fc_b,
                        const float* __restrict__ pooled,
                        float* __restrict__ out) {
  int b = blockIdx.x * blockDim.x + threadIdx.x;
  if (b >= BATCH) return;
  float K = fc_b[0];
  const float invN = 1.0f / (float)N_NODES;
#pragma unroll 8
  for (int d = 0; d < DFEAT; ++d)
    K += (pooled[d] * invN) * fc_w[NCLS + d];
  float logit = K;
#pragma unroll
  for (int c = 0; c < NCLS; ++c)
    logit += vanilla[b * NCLS + c] * fc_w[c];
  const float x = SP_BETA * logit;
  const float sp = fmaxf(x, 0.f) + log1pf(expf(-fabsf(x)));  // stable softplus
  const float t = sp / SP_BETA;
#pragma unroll
  for (int c = 0; c < NCLS; ++c)
    out[b * NCLS + c] = vanilla[b * NCLS + c] / t;
}

// ---------------------------------------------------------------------------
extern "C" void kernel_launch(void* const* d_in, const int* in_sizes, int n_in,
                              void* d_out, int out_size, void* d_ws, size_t ws_size,
                              hipStream_t stream) {
  const float* num_x   = (const float*)d_in[0];   // [99996,128]
  const float* cat_x   = (const float*)d_in[1];   // [4,128,10]
  const float* ew      = (const float*)d_in[2];   // [E]
  const float* vanilla = (const float*)d_in[3];   // [4096,10]
  const float* conv_w  = (const float*)d_in[4];   // [128,128]
  const float* emb_w   = (const float*)d_in[5];   // [4,10]
  const float* fc_w    = (const float*)d_in[6];   // [138]
  const float* fc_b    = (const float*)d_in[7];   // [1]
  const int*   eidx    = (const int*)d_in[8];     // [2,E]
  float* out = (float*)d_out;

  // workspace layout (floats): xw | h | deg/dis | catf | pooled
  float* xw     = (float*)d_ws;
  float* h      = xw + (size_t)N_NODES * DFEAT;
  float* deg    = h + (size_t)N_NODES * DFEAT;
  float* catf   = deg + N_NODES;
  float* pooled = catf + N_CAT * DFEAT;

  const int* rowi = eidx;
  const int* coli = eidx + E_EDGES;

  k_init <<<(N_NODES + 255) / 256, 256, 0, stream>>>(cat_x, emb_w, deg, catf, pooled);
  k_deg  <<<(E_EDGES + 255) / 256, 256, 0, stream>>>(coli, ew, deg);
  k_dis  <<<(N_NODES + 255) / 256, 256, 0, stream>>>(deg);
  k_gemm <<<N_NODES / 16, 256, 0, stream>>>(num_x, catf, conv_w, xw);
  k_self <<<2048, 256, 0, stream>>>(xw, deg, h);
  k_edges<<<4096, 256, 0, stream>>>(rowi, coli, ew, deg, xw, h);
  k_pool <<<256, 256, 0, stream>>>(h, pooled);
  k_final<<<(BATCH + 255) / 256, 256, 0, stream>>>(vanilla, fc_w, fc_b, pooled, out);
}